// GCNLayerV3_14448269984569
// MI455X (gfx1250) — compile-verified
//
#include <hip/hip_runtime.h>
#include <hip/hip_bf16.h>

// CDNA5 (gfx1250) GCN layer: y = x@W via V_WMMA_F32_16X16X4_F32, then
// wave-per-edge SpMM scatter with native global_atomic_add_f32.
// d_ws holds the intermediate y [N_NODES x 128] fp32 (51.2 MB).

typedef __attribute__((ext_vector_type(2))) float v2f;
typedef __attribute__((ext_vector_type(8))) float v8f;

#define DDIM 128
// W is stored in LDS interleaved by K-pairs:
//   Wlds[(k>>1)*KP_STRIDE + n*2 + (k&1)]
// so the B fragment {W[kb][n], W[kb+1][n]} is one contiguous float2 ->
// single ds_load_b64 into an even-aligned VGPR pair (no v_mov shuffles).
// KP_STRIDE = 256 + 32: half-wave K-pair offset == 32 banks (mod 64) ->
// lanes 0-15 use banks {0..30 even}, lanes 16-31 use {32..62 even}: conflict-free.
#define KP_STRIDE 288

// ---------------------------------------------------------------------------
// Kernel 1: y = x @ W.  Block = 256 threads = 8 wave32.
// Each wave computes a 16(M) x 128(N) slab: 8 accumulators of 16x16 f32,
// K swept in steps of 4 with v_wmma_f32_16x16x4_f32 (exact fp32 math).
// ---------------------------------------------------------------------------
__global__ __launch_bounds__(256) void gemm_xw_wmma(
    const float* __restrict__ x, const float* __restrict__ W,
    float* __restrict__ y, int nrows) {
  __shared__ float Wlds[(DDIM / 2) * KP_STRIDE];   // 72 KB

  const int t = threadIdx.x;

  // Cooperative load of W (128x128 f32) into K-pair-interleaved LDS.
  // Coalesced float4 global loads; 4 scattered b32 LDS stores each (one-time).
  for (int idx = t; idx < DDIM * 32; idx += 256) {
    const int r = idx >> 5;        // K row 0..127
    const int q = idx & 31;        // float4 within row
    const float4 w4 = *(const float4*)(W + r * DDIM + q * 4);
    float* dst = &Wlds[(r >> 1) * KP_STRIDE + (q * 4) * 2 + (r & 1)];
    dst[0] = w4.x; dst[2] = w4.y; dst[4] = w4.z; dst[6] = w4.w;
  }
  __syncthreads();

  const int wave = t >> 5;         // 0..7
  const int lane = t & 31;
  const int lh   = lane >> 4;      // half-wave select (0/1)
  const int lm   = lane & 15;

  const int rowBase = blockIdx.x * 128 + wave * 16;
  // Clamp A-row for loads so EXEC stays all-ones through the WMMA region;
  // out-of-range rows are simply never stored.
  const int arow = min(rowBase + lm, nrows - 1);
  const float* __restrict__ xrow = x + (size_t)arow * DDIM;

  v8f acc[8];
#pragma unroll
  for (int nt = 0; nt < 8; ++nt) acc[nt] = (v8f){0.f,0.f,0.f,0.f,0.f,0.f,0.f,0.f};

  // K loop: 32 steps of K=4.
  // A 16x4 f32 frag: lane holds {A[m][kb], A[m][kb+1]}, kb = k0 + 2*lh
  //   -> contiguous float2, global_load_b64.
  // B 4x16 f32 frag: {W[kb][n], W[kb+1][n]} contiguous in interleaved LDS
  //   -> single ds_load_b64, per-nt immediate offset of 32 floats (128 B).
  for (int k0 = 0; k0 < DDIM; k0 += 4) {
    const v2f a = *(const v2f*)(xrow + k0 + 2 * lh);
    const float* __restrict__ wbase =
        &Wlds[((k0 >> 1) + lh) * KP_STRIDE + lm * 2];
#pragma unroll
    for (int nt = 0; nt < 8; ++nt) {
      const v2f b = *(const v2f*)(wbase + nt * 32);
      acc[nt] = __builtin_amdgcn_wmma_f32_16x16x4_f32(
          /*neg_a=*/false, a, /*neg_b=*/false, b,
          /*c_mod=*/(short)0, acc[nt], /*reuse_a=*/false, /*reuse_b=*/false);
    }
  }

  // C/D layout: VGPR r, lanes 0-15 -> M=r, lanes 16-31 -> M=r+8; N = lane%16.
#pragma unroll
  for (int nt = 0; nt < 8; ++nt) {
#pragma unroll
    for (int r = 0; r < 8; ++r) {
      const int m = rowBase + r + 8 * lh;
      if (m < nrows) y[(size_t)m * DDIM + nt * 16 + lm] = acc[nt][r];
    }
  }
}

// ---------------------------------------------------------------------------
// Kernel 2: out[i][:] = bias (float4 stores).  Atomics accumulate on top.
// ---------------------------------------------------------------------------
__global__ __launch_bounds__(256) void bias_init(
    const float* __restrict__ bias, float* __restrict__ out, int total4) {
  const int i = blockIdx.x * 256 + threadIdx.x;
  if (i < total4) {
    const float4 b4 = *(const float4*)(bias + (i & 31) * 4);  // 128/4 = 32 f4/row
    *(float4*)(out + (size_t)i * 4) = b4;
  }
}

// ---------------------------------------------------------------------------
// Kernel 3: SpMM scatter. One wave32 per edge: float4 gather of y[col[e]],
// scale, native f32 atomic add into out[row[e]].  y and out are L2-resident
// (51 MB each vs 192 MB L2), so this runs at L2/atomic-unit bandwidth.
// ---------------------------------------------------------------------------
__global__ __launch_bounds__(256) void spmm_scatter(
    const float* __restrict__ y, const float* __restrict__ ev,
    const int* __restrict__ row, const int* __restrict__ col,
    float* __restrict__ out, int nedges) {
  const int t = threadIdx.x;
  const int wave = t >> 5;
  const int lane = t & 31;
  const int e = blockIdx.x * 8 + wave;
  if (e >= nedges) return;

  const float v = ev[e];
  const int r = row[e];
  const int c = col[e];

  const float4 yv = *(const float4*)(y + (size_t)c * DDIM + lane * 4);
  float* dst = out + (size_t)r * DDIM + lane * 4;
  unsafeAtomicAdd(dst + 0, v * yv.x);
  unsafeAtomicAdd(dst + 1, v * yv.y);
  unsafeAtomicAdd(dst + 2, v * yv.z);
  unsafeAtomicAdd(dst + 3, v * yv.w);
}

// ---------------------------------------------------------------------------
extern "C" void kernel_launch(void* const* d_in, const int* in_sizes, int n_in,
                              void* d_out, int out_size, void* d_ws, size_t ws_size,
                              hipStream_t stream) {
  const float* x    = (const float*)d_in[0];   // [N, 128]
  const float* W    = (const float*)d_in[1];   // [128, 128]
  const float* bias = (const float*)d_in[2];   // [128]
  const float* ev   = (const float*)d_in[3];   // [E]
  const int*   row  = (const int*)d_in[4];     // [E]
  const int*   col  = (const int*)d_in[5];     // [E]
  float* out = (float*)d_out;                  // [N, 128]
  float* y   = (float*)d_ws;                   // [N, 128] scratch (51.2 MB)

  const int n = in_sizes[0] / DDIM;            // 100000
  const int e = in_sizes[3];                   // 640000

  // 1) y = x @ W   (WMMA fp32)
  const int gemm_blocks = (n + 127) / 128;
  gemm_xw_wmma<<<gemm_blocks, 256, 0, stream>>>(x, W, y, n);

  // 2) out = bias broadcast
  const int total4 = n * (DDIM / 4);
  bias_init<<<(total4 + 255) / 256, 256, 0, stream>>>(bias, out, total4);

  // 3) out += scatter(edge_val * y[col]) by row
  spmm_scatter<<<(e + 7) / 8, 256, 0, stream>>>(y, ev, row, col, out, e);
}